// PointLinkTrainer_13134009991767
// MI455X (gfx1250) — compile-verified
//
#include <hip/hip_runtime.h>

#define GRID_N 14
#define NBOX 32

typedef __attribute__((ext_vector_type(2))) float v2f;
typedef __attribute__((ext_vector_type(4))) float v4f;
typedef __attribute__((ext_vector_type(8))) float v8f;

// out_four per image: [4, 196, 4, 51] = 159936 f32 = 3136 rows of 51.
// Tile: 224 rows (224*51 = 11424 f32 = 2856 float4 = 45.7KB), 14 tiles/image.
#define ROWS_PER_TILE 224
#define TILE_F32      11424
#define TILE_V4       2856
#define NTILES        14

__launch_bounds__(256)
__global__ void pointlink_loss_kernel(const float* __restrict__ out4,
                                      const float* __restrict__ bboxes,
                                      const int*   __restrict__ labels,
                                      const int*   __restrict__ Wp,
                                      float*       __restrict__ out) {
  __shared__ float sh_tile[TILE_F32];
  __shared__ int   sh_pix[128];
  __shared__ int   sh_piy[128];
  __shared__ float sh_pdx[128];
  __shared__ float sh_pdy[128];
  __shared__ int   sh_cix[NBOX];
  __shared__ int   sh_ciy[NBOX];
  __shared__ float sh_cdx[NBOX];
  __shared__ float sh_cdy[NBOX];
  __shared__ int   sh_cls[NBOX];
  __shared__ int   sh_wc[196];
  __shared__ int   sh_wz[196];
  __shared__ float sh_red[4 * 256];
  __shared__ float sh_fin[8];

  const int tid = threadIdx.x;
  const int img = blockIdx.x;

  // ---- per-image metadata in LDS ----
  if (tid < 196) { sh_wc[tid] = 128; sh_wz[tid] = NBOX; }
  if (tid < NBOX) {
    const float s = (float)GRID_N / (float)(*Wp);
    const float* bb = bboxes + ((size_t)img * NBOX + tid) * 4;
    float x0 = bb[0] * s, y0 = bb[1] * s, w = bb[2] * s, h = bb[3] * s;
    float px[4] = {x0, x0 + w, x0, x0 + w};
    float py[4] = {y0, y0, y0 + h, y0 + h};
#pragma unroll
    for (int k = 0; k < 4; ++k) {
      float fx = fminf(fmaxf(floorf(px[k]), 0.f), (float)(GRID_N - 1));
      float fy = fminf(fmaxf(floorf(py[k]), 0.f), (float)(GRID_N - 1));
      sh_pix[tid * 4 + k] = (int)fx;
      sh_piy[tid * 4 + k] = (int)fy;
      sh_pdx[tid * 4 + k] = px[k] - fx;
      sh_pdy[tid * 4 + k] = py[k] - fy;
    }
    float cx = x0 + 0.5f * w, cy = y0 + 0.5f * h;
    float fcx = fminf(fmaxf(floorf(cx), 0.f), (float)(GRID_N - 1));
    float fcy = fminf(fmaxf(floorf(cy), 0.f), (float)(GRID_N - 1));
    sh_cix[tid] = (int)fcx; sh_ciy[tid] = (int)fcy;
    sh_cdx[tid] = cx - fcx; sh_cdy[tid] = cy - fcy;
    sh_cls[tid] = labels[(size_t)img * NBOX + tid];
  }
  __syncthreads();
  if (tid < 128)  atomicMin(&sh_wc[sh_pix[tid] * GRID_N + sh_piy[tid]], tid);
  if (tid < NBOX) atomicMin(&sh_wz[sh_cix[tid] * GRID_N + sh_ciy[tid]], tid);
  __syncthreads();

  const float* ibase = out4 + (size_t)img * 159936u;
  float a0 = 0.f, a1 = 0.f, a2 = 0.f, a3 = 0.f;

  for (int tile = 0; tile < NTILES; ++tile) {
    // ---- Phase A: coalesced non-temporal b128 stream into LDS (stride kept) ----
    {
      const v4f* gsrc = (const v4f*)(ibase + tile * TILE_F32);
      v4f* dst = (v4f*)sh_tile;
      for (unsigned e4 = (unsigned)tid; e4 < (unsigned)TILE_V4; e4 += 256u)
        dst[e4] = __builtin_nontemporal_load(gsrc + e4);
    }
    __syncthreads();

    // ---- Phase B: one thread per row; compile-time weighted row scan + sparse fix ----
    if (tid < ROWS_PER_TILE) {
      int R    = tile * ROWS_PER_TILE + tid;      // global row in [0,3136)
      int g    = R / 784;                          // 784 = 196*4
      int rem  = R - g * 784;
      int cell = rem >> 2;
      int slot = rem & 3;

      int   masked, cls, ix, iy;
      float tx, ty;
      if (slot < 2) {                              // corner branch
        int wcv = sh_wc[cell];
        masked  = wcv < 128;
        int ci  = wcv < 128 ? wcv : 0;
        int box = ci >> 2;
        cls = sh_cls[box];
        tx  = sh_pdx[ci];  ty = sh_pdy[ci];
        ix  = sh_cix[box]; iy = sh_ciy[box];
      } else {                                     // center branch
        int wzv = sh_wz[cell];
        masked  = wzv < NBOX;
        int box = wzv < NBOX ? wzv : 0;
        cls = sh_cls[box];
        tx  = sh_cdx[box]; ty = sh_cdy[box];
        ix  = sh_pix[box * 4 + g]; iy = sh_piy[box * 4 + g];
      }

      const float* row = sh_tile + tid * 51;
      float q0  = row[0];
      float q21 = row[21];
      float q22 = row[22];
      float sc = 0.f, sl = 0.f, sy = 0.f;
#pragma unroll
      for (int j = 1; j <= 20; ++j)  { float q = row[j]; sc += q * q; }
#pragma unroll
      for (int j = 23; j <= 36; ++j) { float q = row[j]; sl += q * q; }
#pragma unroll
      for (int j = 37; j <= 50; ++j) { float q = row[j]; sy += q * q; }

      const float w_cls = 1.f / 20.f, w_crd = 0.5f, w_lnk = 1.f / 14.f;
      float Sw = q0 * q0 + w_cls * sc + w_crd * (q21 * q21 + q22 * q22)
               + w_lnk * (sl + sy);

      float pc  = row[1 + cls];
      float plx = row[23 + ix];
      float ply = row[37 + iy];
      float dot = q0 + w_cls * pc + w_crd * (q21 * tx + q22 * ty)
                + w_lnk * (plx + ply);
      float C   = 1.f + w_cls + w_crd * (tx * tx + ty * ty) + 2.f * w_lnk;

      float rt = masked ? (Sw - 2.f * dot + C) : (q0 * q0);
      a0 += (g == 0) ? rt : 0.f;
      a1 += (g == 1) ? rt : 0.f;
      a2 += (g == 2) ? rt : 0.f;
      a3 += (g == 3) ? rt : 0.f;
    }
    __syncthreads();   // before next tile overwrites sh_tile
  }

  sh_red[0 * 256 + tid] = a0;
  sh_red[1 * 256 + tid] = a1;
  sh_red[2 * 256 + tid] = a2;
  sh_red[3 * 256 + tid] = a3;
  __syncthreads();

  // ---- final cross-lane reduction on wave 0 via the matrix unit ----
  // A(16x4) = 64 pre-summed partials, B(4x16) = ones -> D rows are row-sums.
  // Total = sum_M D[M, col]; lane 0 holds M=0..7, lane 16 holds M=8..15.
  // Layout-agnostic for A/B since B == 1 everywhere.
  if (tid < 32) {
#pragma unroll
    for (int g = 0; g < 4; ++g) {
      float s0 = 0.f, s1 = 0.f;
#pragma unroll
      for (int i = 0; i < 4; ++i) s0 += sh_red[g * 256 + tid * 8 + i];
#pragma unroll
      for (int i = 4; i < 8; ++i) s1 += sh_red[g * 256 + tid * 8 + i];
      v2f a; a[0] = s0;  a[1] = s1;
      v2f b; b[0] = 1.f; b[1] = 1.f;
      v8f cacc = {};
      cacc = __builtin_amdgcn_wmma_f32_16x16x4_f32(
          /*neg_a=*/false, a, /*neg_b=*/false, b,
          /*c_mod=*/(short)0, cacc, /*reuse_a=*/false, /*reuse_b=*/false);
      float rs = cacc[0] + cacc[1] + cacc[2] + cacc[3] +
                 cacc[4] + cacc[5] + cacc[6] + cacc[7];
      if (tid == 0)  sh_fin[g * 2 + 0] = rs;   // rows 0..7
      if (tid == 16) sh_fin[g * 2 + 1] = rs;   // rows 8..15
    }
  }
  __syncthreads();
  if (tid < 4)
    out[(size_t)img * 4 + tid] = sh_fin[tid * 2] + sh_fin[tid * 2 + 1];
}

extern "C" void kernel_launch(void* const* d_in, const int* in_sizes, int n_in,
                              void* d_out, int out_size, void* d_ws, size_t ws_size,
                              hipStream_t stream) {
  (void)in_sizes; (void)n_in; (void)d_ws; (void)ws_size; (void)out_size;
  const float* out4 = (const float*)d_in[0];
  const float* bb   = (const float*)d_in[1];
  const int*   lab  = (const int*)d_in[2];
  const int*   Wp   = (const int*)d_in[4];   // inputs: out_four, bboxes, labels, H, W
  float* o = (float*)d_out;
  pointlink_loss_kernel<<<512, 256, 0, stream>>>(out4, bb, lab, Wp, o);
}